// LinearAttention_57853209477816
// MI455X (gfx1250) — compile-verified
//
#include <hip/hip_runtime.h>
#include <hip/hip_bf16.h>

typedef float v2f __attribute__((ext_vector_type(2)));
typedef float v8f __attribute__((ext_vector_type(8)));

#define N_   4
#define L_   4096
#define H_   16
#define D_   64
#define M_   64
#define C_   128
#define NC_  32
#define LDW  68   // LDS row stride (floats): conflict-free column reads, 16B aligned
#define SCW  20   // per-wave scores scratch row stride
#define EPS_ 1e-6f

__device__ __forceinline__ float phi_f(float x) {
    // elu(x)+1 == x+1 (x>0), exp(x) (x<=0)
    return x > 0.f ? x + 1.f : __expf(x);
}

__device__ __forceinline__ v8f wmma4(v2f a, v2f b, v8f c) {
    // D = A(16x4,f32) * B(4x16,f32) + C(16x16,f32)
    return __builtin_amdgcn_wmma_f32_16x16x4_f32(false, a, false, b, short(0), c,
                                                 false, false);
}

// Async global -> LDS copy, 16B per lane, tracked by ASYNCcnt (CDNA5 path).
// LDS byte address = low 32 bits of the generic pointer (LDS aperture rule).
__device__ __forceinline__ void async_b128(const float* lds_dst, const float* gsrc) {
    unsigned int lo = (unsigned int)(size_t)lds_dst;
    asm volatile("global_load_async_to_lds_b128 %0, %1, off"
                 :: "v"(lo), "v"(gsrc)
                 : "memory");
}
__device__ __forceinline__ void async_wait0() {
    asm volatile("s_wait_asynccnt 0x0" ::: "memory");
}

// ---------------------------------------------------------------------------
// Kernel 1: per-chunk KV = phi(K)^T V  (64x64) and ksum = colsum(phi(K))
// grid = N*NC*H blocks, 256 threads (8 waves)
// ---------------------------------------------------------------------------
__global__ __launch_bounds__(256) void kv_build(const float* __restrict__ Kg,
                                                const float* __restrict__ Vg,
                                                float* __restrict__ wskv,
                                                float* __restrict__ wsks) {
    extern __shared__ float smem[];
    float* Ks = smem;             // [128][LDW]
    float* Vs = smem + C_ * LDW;  // [128][LDW]

    int b = blockIdx.x;
    int h = b % H_;
    int t = b / H_;
    int c = t % NC_;
    int n = t / NC_;
    int tid = threadIdx.x;

    // V: async global->LDS (no transform needed), issued first to overlap
    for (int i = tid; i < C_ * 16; i += 256) {
        int row = i >> 4, c4 = (i & 15) << 2;
        size_t g = ((size_t)(n * L_ + c * C_ + row) * H_ + h) * (size_t)D_ + c4;
        async_b128(Vs + row * LDW + c4, Vg + g);
    }
    // K: VGPR path with phi() applied
    for (int i = tid; i < C_ * 16; i += 256) {
        int row = i >> 4, c4 = (i & 15) << 2;
        size_t g = ((size_t)(n * L_ + c * C_ + row) * H_ + h) * (size_t)D_ + c4;
        float4 k4 = *(const float4*)(Kg + g);
        k4.x = phi_f(k4.x); k4.y = phi_f(k4.y); k4.z = phi_f(k4.z); k4.w = phi_f(k4.w);
        *(float4*)(Ks + row * LDW + c4) = k4;
    }
    async_wait0();
    __syncthreads();

    int w    = __builtin_amdgcn_readfirstlane(tid >> 5);
    int lane = tid & 31;
    int ln   = lane & 15;
    int hi   = lane >> 4;
    int kb   = hi << 1;

    // 16 output tiles (4 d-tiles x 4 m-tiles); 2 tiles per wave
    for (int tt = 2 * w; tt < 2 * w + 2; ++tt) {
        int d0 = (tt >> 2) << 4;
        int m0 = (tt & 3) << 4;
        v8f acc = {};
#pragma unroll
        for (int kg = 0; kg < 32; ++kg) {        // K = 128, 4 per WMMA
            int s = 4 * kg + kb;
            v2f a  = { Ks[s * LDW + d0 + ln], Ks[(s + 1) * LDW + d0 + ln] };  // K^T
            v2f bb = { Vs[s * LDW + m0 + ln], Vs[(s + 1) * LDW + m0 + ln] };
            acc = wmma4(a, bb, acc);
        }
        size_t base = ((size_t)(n * NC_ + c) * H_ + h) * (size_t)D_;
#pragma unroll
        for (int j = 0; j < 8; ++j) {
            int d = d0 + j + 8 * hi;
            wskv[(base + d) * M_ + m0 + ln] = acc[j];
        }
    }

    // column sums of phi(K)
    if (tid < D_) {
        float s = 0.f;
        for (int r = 0; r < C_; ++r) s += Ks[r * LDW + tid];
        wsks[((size_t)(n * NC_ + c) * H_ + h) * D_ + tid] = s;
    }
}

// ---------------------------------------------------------------------------
// Kernel 2: exclusive prefix scan over chunks of KV (64x64) and ksum (64)
// grid = N*H blocks, 256 threads
// ---------------------------------------------------------------------------
__global__ __launch_bounds__(256) void kv_scan(float* __restrict__ wskv,
                                               float* __restrict__ wsks) {
    int b = blockIdx.x;
    int n = b / H_, h = b % H_;
    int tid = threadIdx.x;

    float run[16];
#pragma unroll
    for (int j = 0; j < 16; ++j) run[j] = 0.f;
    float runs = 0.f;

    for (int c = 0; c < NC_; ++c) {
        float* p = wskv + ((size_t)(n * NC_ + c) * H_ + h) * (size_t)(D_ * M_);
#pragma unroll
        for (int j = 0; j < 16; ++j) {
            int idx = tid + 256 * j;   // coalesced
            float v = p[idx];
            p[idx]  = run[j];          // write exclusive prefix in place
            run[j] += v;
        }
        if (tid < D_) {
            float* p2 = wsks + ((size_t)(n * NC_ + c) * H_ + h) * D_;
            float v  = p2[tid];
            p2[tid]  = runs;
            runs    += v;
        }
    }
}

// ---------------------------------------------------------------------------
// Kernel 3: out = (Q@KV_prev + tril(Q@K^T)@V) / z
// grid = N*NC*H blocks, 256 threads (8 waves); wave w owns rows 16w..16w+15
// ---------------------------------------------------------------------------
__global__ __launch_bounds__(256) void la_out(const float* __restrict__ Qg,
                                              const float* __restrict__ Kg,
                                              const float* __restrict__ Vg,
                                              const float* __restrict__ wskv,
                                              const float* __restrict__ wsks,
                                              float* __restrict__ Og) {
    extern __shared__ float smem[];
    float* Qs  = smem;
    float* Ks  = Qs  + C_ * LDW;
    float* Vs  = Ks  + C_ * LDW;
    float* KVp = Vs  + C_ * LDW;   // [64][LDW]
    float* ksp = KVp + D_ * LDW;   // [64]
    float* scr = ksp + D_;         // [8 waves][16][SCW]

    int b = blockIdx.x;
    int h = b % H_;
    int t = b / H_;
    int c = t % NC_;
    int n = t / NC_;
    int tid = threadIdx.x;

    // async copies first (V chunk and KV_prev need no transform)
    for (int i = tid; i < C_ * 16; i += 256) {
        int row = i >> 4, c4 = (i & 15) << 2;
        size_t g = ((size_t)(n * L_ + c * C_ + row) * H_ + h) * (size_t)D_ + c4;
        async_b128(Vs + row * LDW + c4, Vg + g);
    }
    size_t kvb = ((size_t)(n * NC_ + c) * H_ + h) * (size_t)(D_ * M_);
    for (int i = tid; i < D_ * 16; i += 256) {
        int row = i >> 4, c4 = (i & 15) << 2;
        async_b128(KVp + row * LDW + c4, wskv + kvb + row * M_ + c4);
    }
    // Q, K through VGPRs with phi()
    for (int i = tid; i < C_ * 16; i += 256) {
        int row = i >> 4, c4 = (i & 15) << 2;
        size_t g = ((size_t)(n * L_ + c * C_ + row) * H_ + h) * (size_t)D_ + c4;
        float4 q4 = *(const float4*)(Qg + g);
        float4 k4 = *(const float4*)(Kg + g);
        q4.x = phi_f(q4.x); q4.y = phi_f(q4.y); q4.z = phi_f(q4.z); q4.w = phi_f(q4.w);
        k4.x = phi_f(k4.x); k4.y = phi_f(k4.y); k4.z = phi_f(k4.z); k4.w = phi_f(k4.w);
        *(float4*)(Qs + row * LDW + c4) = q4;
        *(float4*)(Ks + row * LDW + c4) = k4;
    }
    if (tid < D_)
        ksp[tid] = wsks[((size_t)(n * NC_ + c) * H_ + h) * D_ + tid];
    async_wait0();
    __syncthreads();

    int w    = __builtin_amdgcn_readfirstlane(tid >> 5);
    int lane = tid & 31;
    int ln   = lane & 15;
    int hi   = lane >> 4;
    int kb   = hi << 1;

    // preload Q strip as WMMA A operands (16 K-groups of 4)
    const float* qrow = Qs + (16 * w + ln) * LDW;
    v2f qa[16];
#pragma unroll
    for (int kg = 0; kg < 16; ++kg) {
        int k = 4 * kg + kb;
        v2f a = { qrow[k], qrow[k + 1] };
        qa[kg] = a;
    }

    // inter-chunk: acc = Q_strip @ KV_prev
    v8f acc[4];
#pragma unroll
    for (int nt = 0; nt < 4; ++nt) {
        v8f a_ = {};
#pragma unroll
        for (int kg = 0; kg < 16; ++kg) {
            int k = 4 * kg + kb;
            v2f bb = { KVp[k * LDW + nt * 16 + ln], KVp[(k + 1) * LDW + nt * 16 + ln] };
            a_ = wmma4(qa[kg], bb, a_);
        }
        acc[nt] = a_;
    }

    // z part 1: q . ksum_prev  (full row dot after xor-16 combine)
    float qz = 0.f;
#pragma unroll
    for (int kg = 0; kg < 16; ++kg) {
        int k = 4 * kg + kb;
        qz += qa[kg][0] * ksp[k] + qa[kg][1] * ksp[k + 1];
    }
    qz += __shfl_xor(qz, 16, 32);

    float zs[8];
#pragma unroll
    for (int j = 0; j < 8; ++j) zs[j] = 0.f;

    float* myscr = scr + w * 16 * SCW;

    for (int tt = 0; tt <= w; ++tt) {          // causal t-tiles
        int t0 = tt * 16;
        // scores tile = Q_strip @ K_tile^T
        v8f sc = {};
        const float* krow = Ks + (t0 + ln) * LDW;
#pragma unroll
        for (int kg = 0; kg < 16; ++kg) {
            int k = 4 * kg + kb;
            v2f bb = { krow[k], krow[k + 1] };   // B[k][t] = K[t][k], contiguous
            sc = wmma4(qa[kg], bb, sc);
        }
        // mask diagonal tile, accumulate row-sums into z, stage tile in LDS
#pragma unroll
        for (int j = 0; j < 8; ++j) {
            int r = j + 8 * hi;
            float v = sc[j];
            if (tt == w) v = (ln <= r) ? v : 0.f;
            float rs = v;
            rs += __shfl_xor(rs, 1, 32);
            rs += __shfl_xor(rs, 2, 32);
            rs += __shfl_xor(rs, 4, 32);
            rs += __shfl_xor(rs, 8, 32);
            zs[j] += rs;
            myscr[r * SCW + ln] = v;
        }
        // intra: acc += scores_tile @ V_tile   (re-read tile in A layout)
#pragma unroll
        for (int kg2 = 0; kg2 < 4; ++kg2) {
            int k = 4 * kg2 + kb;
            v2f a2 = { myscr[ln * SCW + k], myscr[ln * SCW + k + 1] };
#pragma unroll
            for (int nt = 0; nt < 4; ++nt) {
                v2f bb = { Vs[(t0 + k) * LDW + nt * 16 + ln],
                           Vs[(t0 + k + 1) * LDW + nt * 16 + ln] };
                acc[nt] = wmma4(a2, bb, acc[nt]);
            }
        }
    }

    // divide by z and store
    size_t ob = (size_t)(n * L_ + c * C_ + 16 * w);
#pragma unroll
    for (int j = 0; j < 8; ++j) {
        int row = j + 8 * hi;
        float zr  = __shfl(qz, row, 32) + zs[j] + EPS_;
        float inv = 1.f / zr;
        size_t o = ((ob + row) * H_ + h) * (size_t)M_;
#pragma unroll
        for (int nt = 0; nt < 4; ++nt)
            Og[o + nt * 16 + ln] = acc[nt][j] * inv;
    }
}

// ---------------------------------------------------------------------------
extern "C" void kernel_launch(void* const* d_in, const int* in_sizes, int n_in,
                              void* d_out, int out_size, void* d_ws, size_t ws_size,
                              hipStream_t stream) {
    const float* Q = (const float*)d_in[0];
    const float* K = (const float*)d_in[1];
    const float* V = (const float*)d_in[2];
    float* O = (float*)d_out;

    float* wskv = (float*)d_ws;                                   // [N][NC][H][64][64]
    float* wsks = wskv + (size_t)N_ * NC_ * H_ * D_ * M_;         // [N][NC][H][64]

    dim3 blk(256);
    size_t s1 = (size_t)(2 * C_ * LDW) * sizeof(float);
    kv_build<<<dim3(N_ * NC_ * H_), blk, s1, stream>>>(K, V, wskv, wsks);

    kv_scan<<<dim3(N_ * H_), blk, 0, stream>>>(wskv, wsks);

    size_t s3 = (size_t)(3 * C_ * LDW + D_ * LDW + D_ + 8 * 16 * SCW) * sizeof(float);
    la_out<<<dim3(N_ * NC_ * H_), blk, s3, stream>>>(Q, K, V, wskv, wsks, O);
}